// DiffeomorphismNet_10969346474226
// MI455X (gfx1250) — compile-verified
//
#include <hip/hip_runtime.h>
#include <hip/hip_bf16.h>
#include <math.h>

// ---------------------------------------------------------------------------
// DiffeomorphismNet on MI455X (gfx1250), fp32 WMMA (V_WMMA_F32_16X16X4_F32).
//
//   out[:,   0:128] = h_out      (forward MLP on xt)
//   out[:, 128:256] = h_dot      (JVP of Jacobian chain with xdot)
//   out[:, 256:384] = zero_norm  (row norms of Gz[:, :, :128])
//
// GEMM inner loops use explicit multi-stage software pipelining with DISTINCT
// buffer sets (no inter-set copies). The hot zero_chain kernel (146 GFLOP,
// 1 WG/WGP due to 257 KB LDS) uses a 4-stage pipeline for ~3 stages of load
// latency cover; the occupancy-rich forward GEMMs use 2 stages.
// ---------------------------------------------------------------------------

typedef float v2f __attribute__((ext_vector_type(2)));
typedef float v8f __attribute__((ext_vector_type(8)));

#define HID   1024
#define NST   128
#define BATCH 256
#define IN2   256    // 2n
#define CB    32     // Jacobian columns per zero-chain workgroup
#define TS    1028   // LDS tile row stride (1024 + 4 pad -> stride-4 banks)

__device__ __forceinline__ v8f wmma_f32(v2f a, v2f b, v8f c) {
  // D = A(16x4 f32) * B(4x16 f32) + C(16x16 f32)
  return __builtin_amdgcn_wmma_f32_16x16x4_f32(
      /*neg_a=*/false, a, /*neg_b=*/false, b,
      /*c_mod=*/(short)0, c, /*reuse_a=*/false, /*reuse_b=*/false);
}

// ---------------------------------------------------------------------------
// X2[0:256]   = xt     = x[:, 0:256]
// X2[256:512] = x_zero = concat(x[:,384:512], x[:,128:256])
// ---------------------------------------------------------------------------
__global__ void build_x2(const float* __restrict__ x, float* __restrict__ X2) {
  int i = blockIdx.x * blockDim.x + threadIdx.x;
  if (i >= 512 * IN2) return;
  int row = i >> 8, c = i & 255;
  float v;
  if (row < 256) {
    v = x[row * 512 + c];
  } else {
    int rz = row - 256;
    v = (c < 128) ? x[rz * 512 + 384 + c] : x[rz * 512 + 128 + (c - 128)];
  }
  X2[i] = v;
}

__global__ void zero_buf(float* __restrict__ p, int n) {
  int i = blockIdx.x * blockDim.x + threadIdx.x;
  if (i < n) p[i] = 0.f;
}

// ---------------------------------------------------------------------------
// C[M,NN] = A[M,K] @ W[NN,K]^T  (+bias)  (epilogue: 0=none 1=relu 2=sign)
// then optional elementwise multiply by mask[M,NN].
// Block: 256 threads = 8 waves; block tile 128(M) x 64(N); wave tile 16x64.
// f32 WMMA A/B frag layout: lane L -> row base + (L&15), K = (L>>4)*2 + {0,1}.
// Requires K % 8 == 0.
// ---------------------------------------------------------------------------
__global__ __launch_bounds__(256) void gemm_awt(
    const float* __restrict__ A, int lda,
    const float* __restrict__ W, int ldw,
    const float* __restrict__ bias,
    const float* __restrict__ mask, int ldm,
    float* __restrict__ C, int ldc,
    int M, int K, int epi)
{
  const int lane = threadIdx.x & 31, wv = threadIdx.x >> 5;
  const int l16 = lane & 15, hlf = lane >> 4;
  const int m0 = blockIdx.x * 128 + wv * 16;
  const int n0 = blockIdx.y * 64;
  if (m0 >= M) return;  // wave-uniform

  const v8f vz = {0.f,0.f,0.f,0.f,0.f,0.f,0.f,0.f};
  v8f acc[4] = {vz, vz, vz, vz};

  const int kof = hlf * 2;
  const float* Ar  = A + (size_t)(m0 + l16) * lda;
  const float* Wr0 = W + (size_t)(n0 +  0 + l16) * ldw;
  const float* Wr1 = W + (size_t)(n0 + 16 + l16) * ldw;
  const float* Wr2 = W + (size_t)(n0 + 32 + l16) * ldw;
  const float* Wr3 = W + (size_t)(n0 + 48 + l16) * ldw;

  // two distinct fragment sets (no copies between them)
  v2f a0, b00, b10, b20, b30;   // set 0
  v2f a1, b01, b11, b21, b31;   // set 1

#define LDSET0(kk) { a0  = *(const v2f*)(Ar  + (kk)); \
                     b00 = *(const v2f*)(Wr0 + (kk)); \
                     b10 = *(const v2f*)(Wr1 + (kk)); \
                     b20 = *(const v2f*)(Wr2 + (kk)); \
                     b30 = *(const v2f*)(Wr3 + (kk)); }
#define LDSET1(kk) { a1  = *(const v2f*)(Ar  + (kk)); \
                     b01 = *(const v2f*)(Wr0 + (kk)); \
                     b11 = *(const v2f*)(Wr1 + (kk)); \
                     b21 = *(const v2f*)(Wr2 + (kk)); \
                     b31 = *(const v2f*)(Wr3 + (kk)); }
#define WM0() { acc[0] = wmma_f32(a0, b00, acc[0]); \
                acc[1] = wmma_f32(a0, b10, acc[1]); \
                acc[2] = wmma_f32(a0, b20, acc[2]); \
                acc[3] = wmma_f32(a0, b30, acc[3]); }
#define WM1() { acc[0] = wmma_f32(a1, b01, acc[0]); \
                acc[1] = wmma_f32(a1, b11, acc[1]); \
                acc[2] = wmma_f32(a1, b21, acc[2]); \
                acc[3] = wmma_f32(a1, b31, acc[3]); }

  LDSET0(kof);
  LDSET1(kof + 4);
  for (int k = 0; k < K - 8; k += 8) {
    WM0(); LDSET0(k + 8  + kof);
    WM1(); LDSET1(k + 12 + kof);
  }
  WM0();
  WM1();
#undef LDSET0
#undef LDSET1
#undef WM0
#undef WM1

  // C/D frag: vgpr r -> row m0 + r + 8*(lane>=16), col n0 + nt*16 + (lane&15)
  #pragma unroll
  for (int nt = 0; nt < 4; ++nt) {
    #pragma unroll
    for (int r = 0; r < 8; ++r) {
      const int m  = m0 + r + 8 * hlf;
      const int nn = n0 + nt * 16 + l16;
      float v = acc[nt][r];
      if (bias) v += bias[nn];
      if (epi == 1)      v = v > 0.f ? v   : 0.f;
      else if (epi == 2) v = v > 0.f ? 1.f : 0.f;
      if (mask) v *= mask[(size_t)m * ldm + nn];
      C[(size_t)m * ldc + nn] = v;
    }
  }
}

// ---------------------------------------------------------------------------
// C[M,N] = A[M,K] @ B[K,N]  (B row-major, N-contiguous) -- M0 = Wh0 @ W_in[:,:128]
// ---------------------------------------------------------------------------
__global__ __launch_bounds__(256) void gemm_ab(
    const float* __restrict__ A, int lda,
    const float* __restrict__ B, int ldb,
    float* __restrict__ C, int ldc,
    int M, int K)
{
  const int lane = threadIdx.x & 31, wv = threadIdx.x >> 5;
  const int l16 = lane & 15, hlf = lane >> 4;
  const int m0 = blockIdx.x * 128 + wv * 16;
  const int n0 = blockIdx.y * 64;
  if (m0 >= M) return;

  const v8f vz = {0.f,0.f,0.f,0.f,0.f,0.f,0.f,0.f};
  v8f acc[4] = {vz, vz, vz, vz};
  const int kof = hlf * 2;
  const float* Ar = A + (size_t)(m0 + l16) * lda;
  const float* Bc = B + n0 + l16;

  v2f a0, c00, c10, c20, c30;   // set 0
  v2f a1, c01, c11, c21, c31;   // set 1

#define BLD(dst, kk, nt) { (dst).x = Bc[(size_t)(kk) * ldb + (nt) * 16]; \
                           (dst).y = Bc[(size_t)((kk) + 1) * ldb + (nt) * 16]; }
#define LDSET0(kk) { a0 = *(const v2f*)(Ar + (kk)); \
                     BLD(c00, (kk), 0) BLD(c10, (kk), 1) BLD(c20, (kk), 2) BLD(c30, (kk), 3) }
#define LDSET1(kk) { a1 = *(const v2f*)(Ar + (kk)); \
                     BLD(c01, (kk), 0) BLD(c11, (kk), 1) BLD(c21, (kk), 2) BLD(c31, (kk), 3) }
#define WM0() { acc[0] = wmma_f32(a0, c00, acc[0]); acc[1] = wmma_f32(a0, c10, acc[1]); \
                acc[2] = wmma_f32(a0, c20, acc[2]); acc[3] = wmma_f32(a0, c30, acc[3]); }
#define WM1() { acc[0] = wmma_f32(a1, c01, acc[0]); acc[1] = wmma_f32(a1, c11, acc[1]); \
                acc[2] = wmma_f32(a1, c21, acc[2]); acc[3] = wmma_f32(a1, c31, acc[3]); }

  LDSET0(kof);
  LDSET1(kof + 4);
  for (int k = 0; k < K - 8; k += 8) {
    WM0(); LDSET0(k + 8  + kof);
    WM1(); LDSET1(k + 12 + kof);
  }
  WM0();
  WM1();
#undef BLD
#undef LDSET0
#undef LDSET1
#undef WM0
#undef WM1

  #pragma unroll
  for (int nt = 0; nt < 4; ++nt)
    #pragma unroll
    for (int r = 0; r < 8; ++r)
      C[(size_t)(m0 + r + 8 * hlf) * ldc + n0 + nt * 16 + l16] = acc[nt][r];
}

// ---------------------------------------------------------------------------
// Fragment loaders / WMMA burst for the Jacobian-chain GEMM (inlined; arrays
// fully unrolled so everything stays in registers).
// ---------------------------------------------------------------------------
__device__ __forceinline__ void chain_ld(
    const float* __restrict__ Wr, const float* __restrict__ Tb0,
    const float* __restrict__ Tb1, int kk,
    v2f av[8], v2f& c0, v2f& c1)
{
  c0 = *(const v2f*)(Tb0 + kk);
  c1 = *(const v2f*)(Tb1 + kk);
  #pragma unroll
  for (int mt = 0; mt < 8; ++mt)
    av[mt] = *(const v2f*)(Wr + (size_t)(mt * 16 * HID) + kk);
}

__device__ __forceinline__ void chain_mm(const v2f av[8], v2f c0, v2f c1,
                                         v8f acc[8][2])
{
  #pragma unroll
  for (int mt = 0; mt < 8; ++mt) {
    acc[mt][0] = wmma_f32(av[mt], c0, acc[mt][0]);
    acc[mt][1] = wmma_f32(av[mt], c1, acc[mt][1]);
  }
}

// ---------------------------------------------------------------------------
// One Jacobian-chain step: Tout = mask ⊙ (Wm[1024,1024] @ Tin).
// LDS tiles are TRANSPOSED: T[c][k], row stride TS=1028 floats, so lane l16
// reads a contiguous v2f (ds_load_b64, distinct stride-4 banks) and the
// masked write-back is conflict-free ds_store_b128s.
// Wave wv owns rows [wv*128, wv*128+128): 16 f32 accumulators; each Wm
// element read exactly once per WG (streams from the 192 MB L2).
// 4-stage pipelined K loop with distinct fragment sets: ~3 stages (~48 WMMAs)
// of load-latency cover, since 257 KB LDS limits us to 1 WG/WGP (2 waves/SIMD)
// and intra-wave overlap is the primary latency-hiding mechanism here.
// ---------------------------------------------------------------------------
__device__ __forceinline__ void chain_gemm_1024(
    const float* __restrict__ Wm,
    const float* __restrict__ Tin,
    float* __restrict__ Tout,
    const float* __restrict__ dm,
    int lane, int wv)
{
  const int l16 = lane & 15, hlf = lane >> 4;
  const int kof = hlf * 2;
  const v8f vz = {0.f,0.f,0.f,0.f,0.f,0.f,0.f,0.f};
  v8f acc[8][2];
  #pragma unroll
  for (int mt = 0; mt < 8; ++mt) { acc[mt][0] = vz; acc[mt][1] = vz; }

  const float* Tb0 = Tin + (size_t)l16 * TS;
  const float* Tb1 = Tin + (size_t)(16 + l16) * TS;
  const float* Wr  = Wm + (size_t)(wv * 128 + l16) * HID;

  // four distinct fragment sets (prefetch distance = 3 pipeline stages)
  v2f avA[8], cA0, cA1;
  v2f avB[8], cB0, cB1;
  v2f avC[8], cC0, cC1;
  v2f avD[8], cD0, cD1;

  chain_ld(Wr, Tb0, Tb1, kof,      avA, cA0, cA1);
  chain_ld(Wr, Tb0, Tb1, kof + 4,  avB, cB0, cB1);
  chain_ld(Wr, Tb0, Tb1, kof + 8,  avC, cC0, cC1);
  chain_ld(Wr, Tb0, Tb1, kof + 12, avD, cD0, cD1);
  for (int k = 0; k < HID - 16; k += 16) {
    chain_mm(avA, cA0, cA1, acc);
    chain_ld(Wr, Tb0, Tb1, k + 16 + kof, avA, cA0, cA1);
    chain_mm(avB, cB0, cB1, acc);
    chain_ld(Wr, Tb0, Tb1, k + 20 + kof, avB, cB0, cB1);
    chain_mm(avC, cC0, cC1, acc);
    chain_ld(Wr, Tb0, Tb1, k + 24 + kof, avC, cC0, cC1);
    chain_mm(avD, cD0, cD1, acc);
    chain_ld(Wr, Tb0, Tb1, k + 28 + kof, avD, cD0, cD1);
  }
  chain_mm(avA, cA0, cA1, acc);
  chain_mm(avB, cB0, cB1, acc);
  chain_mm(avC, cC0, cC1, acc);
  chain_mm(avD, cD0, cD1, acc);

  // masked transposed store: rows contiguous per lane-column -> float4 stores
  #pragma unroll
  for (int mt = 0; mt < 8; ++mt) {
    const int mb = wv * 128 + mt * 16 + 8 * hlf;   // 8 consecutive rows
    const float4 dA = *(const float4*)(dm + mb);
    const float4 dB = *(const float4*)(dm + mb + 4);
    float4 v0l, v0h, v1l, v1h;
    v0l.x = acc[mt][0][0] * dA.x; v0l.y = acc[mt][0][1] * dA.y;
    v0l.z = acc[mt][0][2] * dA.z; v0l.w = acc[mt][0][3] * dA.w;
    v0h.x = acc[mt][0][4] * dB.x; v0h.y = acc[mt][0][5] * dB.y;
    v0h.z = acc[mt][0][6] * dB.z; v0h.w = acc[mt][0][7] * dB.w;
    v1l.x = acc[mt][1][0] * dA.x; v1l.y = acc[mt][1][1] * dA.y;
    v1l.z = acc[mt][1][2] * dA.z; v1l.w = acc[mt][1][3] * dA.w;
    v1h.x = acc[mt][1][4] * dB.x; v1h.y = acc[mt][1][5] * dB.y;
    v1h.z = acc[mt][1][6] * dB.z; v1h.w = acc[mt][1][7] * dB.w;
    *(float4*)(Tout + (size_t)l16 * TS + mb)            = v0l;
    *(float4*)(Tout + (size_t)l16 * TS + mb + 4)        = v0h;
    *(float4*)(Tout + (size_t)(16 + l16) * TS + mb)     = v1l;
    *(float4*)(Tout + (size_t)(16 + l16) * TS + mb + 4) = v1h;
  }
}

// ---------------------------------------------------------------------------
// Zero-branch Jacobian chain. grid = (4 column-blocks, 256 batches),
// 256 threads. LDS: two transposed [CB][TS] f32 tiles (~257 KB < 320 KB WGP).
//   T0 = d1 ⊙ M0[:, c0:c0+32]
//   T1 = d2 ⊙ (Wh1 @ T0);  T0 = d3 ⊙ (Wh2 @ T1)
//   G  = W_out @ T0 [128x32]; per-row sum of squares via 16-lane shfl_xor
//   reduction + global f32 atomics.
// ---------------------------------------------------------------------------
__global__ __launch_bounds__(256) void zero_chain(
    const float* __restrict__ M0,
    const float* __restrict__ Wh1, const float* __restrict__ Wh2,
    const float* __restrict__ d1z, const float* __restrict__ d2z,
    const float* __restrict__ d3z,
    const float* __restrict__ Wout,
    float* __restrict__ zsum)
{
  extern __shared__ float lds[];
  float* T0 = lds;            // [CB][TS]
  float* T1 = lds + CB * TS;  // [CB][TS]

  const int cb = blockIdx.x;      // 0..3
  const int b  = blockIdx.y;      // 0..255
  const int c0 = cb * CB;
  const int lane = threadIdx.x & 31, wv = threadIdx.x >> 5;

  const float* d1 = d1z + (size_t)b * HID;
  const float* d2 = d2z + (size_t)b * HID;
  const float* d3 = d3z + (size_t)b * HID;

  // T0[c][j] = M0[j][c0+c] * d1[j]  (coalesced M0 reads, transposed LDS write)
  for (int idx = threadIdx.x; idx < HID * CB; idx += 256) {
    const int j = idx >> 5, c = idx & 31;
    T0[(size_t)c * TS + j] = M0[(size_t)j * NST + c0 + c] * d1[j];
  }
  __syncthreads();

  chain_gemm_1024(Wh1, T0, T1, d2, lane, wv);
  __syncthreads();
  chain_gemm_1024(Wh2, T1, T0, d3, lane, wv);
  __syncthreads();

  // G = Wout @ T0 : wave wv owns rows wv*16 .. wv*16+15 (2-stage pipelined)
  const int l16 = lane & 15, hlf = lane >> 4;
  const int kof = hlf * 2;
  const v8f vz = {0.f,0.f,0.f,0.f,0.f,0.f,0.f,0.f};
  v8f g0 = vz, g1 = vz;
  const float* Tb0 = T0 + (size_t)l16 * TS;
  const float* Tb1 = T0 + (size_t)(16 + l16) * TS;
  const float* Wr  = Wout + (size_t)(wv * 16 + l16) * HID;

  v2f a0, c00, c10;   // set 0
  v2f a1, c01, c11;   // set 1
#define FLD0(kk) { a0 = *(const v2f*)(Wr + (kk)); \
                   c00 = *(const v2f*)(Tb0 + (kk)); c10 = *(const v2f*)(Tb1 + (kk)); }
#define FLD1(kk) { a1 = *(const v2f*)(Wr + (kk)); \
                   c01 = *(const v2f*)(Tb0 + (kk)); c11 = *(const v2f*)(Tb1 + (kk)); }
  FLD0(kof);
  FLD1(kof + 4);
  for (int k = 0; k < HID - 8; k += 8) {
    g0 = wmma_f32(a0, c00, g0); g1 = wmma_f32(a0, c10, g1);
    FLD0(k + 8 + kof);
    g0 = wmma_f32(a1, c01, g0); g1 = wmma_f32(a1, c11, g1);
    FLD1(k + 12 + kof);
  }
  g0 = wmma_f32(a0, c00, g0); g1 = wmma_f32(a0, c10, g1);
  g0 = wmma_f32(a1, c01, g0); g1 = wmma_f32(a1, c11, g1);
#undef FLD0
#undef FLD1

  #pragma unroll
  for (int r = 0; r < 8; ++r) {
    // row m = wv*16 + r + 8*hlf ; g0 col = l16, g1 col = 16+l16 (same row)
    float s = g0[r] * g0[r] + g1[r] * g1[r];
    s += __shfl_xor(s, 1);
    s += __shfl_xor(s, 2);
    s += __shfl_xor(s, 4);
    s += __shfl_xor(s, 8);
    if (l16 == 0)
      atomicAdd(&zsum[(size_t)b * NST + wv * 16 + r + 8 * hlf], s);
  }
}

__global__ void finalize_norm(const float* __restrict__ zsum,
                              float* __restrict__ out) {
  int i = blockIdx.x * blockDim.x + threadIdx.x;   // BATCH*NST
  if (i >= BATCH * NST) return;
  int b = i >> 7, c = i & 127;
  out[(size_t)b * 384 + 256 + c] = sqrtf(zsum[i]);
}

// ---------------------------------------------------------------------------
extern "C" void kernel_launch(void* const* d_in, const int* in_sizes, int n_in,
                              void* d_out, int out_size, void* d_ws, size_t ws_size,
                              hipStream_t stream)
{
  (void)in_sizes; (void)n_in; (void)out_size; (void)ws_size;
  const float* x     = (const float*)d_in[0];
  const float* W_in  = (const float*)d_in[1];
  const float* b_in  = (const float*)d_in[2];
  const float* Wh    = (const float*)d_in[3];
  const float* bh    = (const float*)d_in[4];
  const float* W_out = (const float*)d_in[5];
  const float* b_out = (const float*)d_in[6];
  float* out = (float*)d_out;
  float* ws  = (float*)d_ws;

  const float* Wh0 = Wh;
  const float* Wh1 = Wh + 1024 * 1024;
  const float* Wh2 = Wh + 2 * 1024 * 1024;
  const float* bh0 = bh, *bh1 = bh + 1024, *bh2 = bh + 2048;

  // workspace layout (floats) ~13.2 MiB total
  float* X2  = ws;                  // 512*256
  float* hA  = X2  + 512 * 256;     // 512*1024
  float* hB  = hA  + 512 * 1024;    // 512*1024
  float* D1  = hB  + 512 * 1024;    // 512*1024
  float* D2  = D1  + 512 * 1024;    // 512*1024
  float* D3  = D2  + 512 * 1024;    // 512*1024
  float* uA  = D3  + 512 * 1024;    // 256*1024
  float* uB  = uA  + 256 * 1024;    // 256*1024
  float* M0b = uB  + 256 * 1024;    // 1024*128
  float* zs  = M0b + 1024 * 128;    // 256*128

  dim3 blk(256);
  build_x2<<<dim3((512 * 256 + 255) / 256), blk, 0, stream>>>(x, X2);
  zero_buf<<<dim3((256 * 128 + 255) / 256), blk, 0, stream>>>(zs, 256 * 128);

  // ---- forward passes (xt rows 0..255, x_zero rows 256..511) + deltas ----
  gemm_awt<<<dim3(4,16), blk, 0, stream>>>(X2,256,  W_in,256, b_in, nullptr,0, hA,1024, 512,256,  0); // h0
  gemm_awt<<<dim3(4,16), blk, 0, stream>>>(hA,1024, Wh0,1024, bh0,  nullptr,0, hB,1024, 512,1024, 1); // h1
  gemm_awt<<<dim3(4,16), blk, 0, stream>>>(hB,1024, Wh0,1024, bh0,  nullptr,0, D1,1024, 512,1024, 2); // d1 (double apply)
  gemm_awt<<<dim3(4,16), blk, 0, stream>>>(hB,1024, Wh1,1024, bh1,  nullptr,0, hA,1024, 512,1024, 1); // h2
  gemm_awt<<<dim3(4,16), blk, 0, stream>>>(hA,1024, Wh1,1024, bh1,  nullptr,0, D2,1024, 512,1024, 2); // d2
  gemm_awt<<<dim3(4,16), blk, 0, stream>>>(hA,1024, Wh2,1024, bh2,  nullptr,0, hB,1024, 512,1024, 1); // h3
  gemm_awt<<<dim3(4,16), blk, 0, stream>>>(hB,1024, Wh2,1024, bh2,  nullptr,0, D3,1024, 512,1024, 2); // d3
  // h_out -> out[:, 0:128]
  gemm_awt<<<dim3(2,2),  blk, 0, stream>>>(hB,1024, W_out,1024, b_out, nullptr,0, out,384, 256,1024, 0);

  // ---- h_dot: JVP through the chain (xt rows only) -> out[:, 128:256] ----
  gemm_awt<<<dim3(2,16), blk, 0, stream>>>(x + 256,512, W_in,256, nullptr, nullptr,0, uA,1024, 256,128,  0);
  gemm_awt<<<dim3(2,16), blk, 0, stream>>>(uA,1024, Wh0,1024, nullptr, D1,1024, uB,1024, 256,1024, 0);
  gemm_awt<<<dim3(2,16), blk, 0, stream>>>(uB,1024, Wh1,1024, nullptr, D2,1024, uA,1024, 256,1024, 0);
  gemm_awt<<<dim3(2,16), blk, 0, stream>>>(uA,1024, Wh2,1024, nullptr, D3,1024, uB,1024, 256,1024, 0);
  gemm_awt<<<dim3(2,2),  blk, 0, stream>>>(uB,1024, W_out,1024, nullptr, nullptr,0, out + 128,384, 256,1024, 0);

  // ---- zero_norm: per-batch Jacobian chain -> out[:, 256:384] ----
  gemm_ab<<<dim3(8,2), blk, 0, stream>>>(Wh0,1024, W_in,256, M0b,128, 1024,1024);
  zero_chain<<<dim3(4,256), blk, 2 * CB * TS * sizeof(float), stream>>>(
      M0b, Wh1, Wh2,
      D1 + 256 * 1024, D2 + 256 * 1024, D3 + 256 * 1024,
      W_out, zs);
  finalize_norm<<<dim3((256 * 128 + 255) / 256), blk, 0, stream>>>(zs, out);
}